// TopKRouter_28741921145174
// MI455X (gfx1250) — compile-verified
//
#include <hip/hip_runtime.h>

typedef float v2f __attribute__((ext_vector_type(2)));
typedef float v8f __attribute__((ext_vector_type(8)));

#define T_TOK 16384
#define H_DIM 4096
#define E_EXP 64
#define TOPK  8
#define NEG_MASK -10000.0f
#define NEG_INF  -3.402823466e38f

#define CK     512             // K-chunk staged in LDS per double-buffer slot
#define NCHUNK (H_DIM / CK)    // 8
#define APAD   4               // +4 floats/row -> conflict-free ds_load_b64
#define AROW   (CK + APAD)     // 516

__device__ __forceinline__ v8f wmma_f32(v2f a, v2f b, v8f c) {
    // 8 args: (neg_a, A, neg_b, B, c_mod, C, reuse_a, reuse_b)
    return __builtin_amdgcn_wmma_f32_16x16x4_f32(
        false, a, false, b, (short)0, c, false, false);
}

// ---------------------------------------------------------------------------
// Kernel 1: router logits GEMM.
//  - Block = 128 threads (4 waves) owns 16 token rows; wave w owns experts
//    [16w, 16w+16). K swept via V_WMMA_F32_16X16X4_F32 with 4 independent
//    accumulator chains (breaks the WMMA C-operand RAW serialization).
//  - A-tile staged in LDS with GLOBAL_LOAD_ASYNC_TO_LDS_B128 (ASYNCcnt DMA),
//    double-buffered over 8 chunks of K=512; cuts the 4x redundant per-wave
//    A reads and exercises the CDNA5 async-to-LDS path.
//  - Per-lane WMMA layouts per CDNA5 ISA 7.12.2:
//      A 16x4 f32 : lane L -> row M=L%16, VGPRs = K = 2*(L/16)+{0,1}
//      B 4x16 f32 : lane L -> col N=L%16, VGPRs = K = 2*(L/16)+{0,1}
//      D 16x16 f32: VGPR i on lane L -> M = i + 8*(L/16), N = L%16
// ---------------------------------------------------------------------------
__global__ __launch_bounds__(128) void router_gemm_kernel(
    const float* __restrict__ hs,          // [T, H]
    const float* __restrict__ wt,          // [E, H]
    const int*   __restrict__ allowed_ids, // [32]
    float*       __restrict__ logits_out)  // [T, E] (masked)
{
    __shared__ float tile[2][16][AROW];

    const int tid        = threadIdx.x;
    const int lane       = tid & 31;
    const int wave       = tid >> 5;
    const int rowBase    = blockIdx.x * 16;
    const int expertBase = wave * 16;

    const int m    = lane & 15;   // A row / B col within tile
    const int half = lane >> 4;   // 0: K pair {0,1}, 1: K pair {2,3}
    const int koff = half * 2;

    const float* __restrict__ aBase = hs + (size_t)rowBase * H_DIM;
    const float* __restrict__ brow  = wt + (size_t)(expertBase + m) * H_DIM;

    // Each of the 128 threads DMAs a 16-byte segment of each of the 16 rows:
    // thread t covers floats [4t, 4t+4) of every row in the chunk.
    auto issue_chunk = [&](int c) {
        const float* g0 = aBase + (size_t)c * CK + 4 * tid;
        // Generic pointer to LDS: low 32 bits are the LDS byte address
        // (aperture mapping).
        uint32_t l0 = (uint32_t)(uintptr_t)&tile[c & 1][0][4 * tid];
#pragma unroll
        for (int i = 0; i < 16; ++i) {
            uint64_t g = (uint64_t)(uintptr_t)(g0 + (size_t)i * H_DIM);
            uint32_t l = l0 + (uint32_t)(i * AROW * 4);
            asm volatile("global_load_async_to_lds_b128 %0, %1, off"
                         :: "v"(l), "v"(g) : "memory");
        }
    };

    v8f acc0 = {}, acc1 = {}, acc2 = {}, acc3 = {};

    issue_chunk(0);
    for (int c = 0; c < NCHUNK; ++c) {
        if (c + 1 < NCHUNK) {
            issue_chunk(c + 1);
            // 16 outstanding = next chunk in flight; this chunk's 16 done.
            asm volatile("s_wait_asynccnt 0x10" ::: "memory");
        } else {
            asm volatile("s_wait_asynccnt 0x0" ::: "memory");
        }
        __syncthreads();   // chunk c fully in LDS across all 4 waves

        const float* __restrict__ arow = &tile[c & 1][m][koff]; // ds_load_b64
        const float* __restrict__ bp   = brow + (size_t)c * CK + koff;
#pragma unroll 4
        for (int k = 0; k < CK; k += 16) {
            v2f a0 = *(const v2f*)(arow + k);
            v2f b0 = *(const v2f*)(bp + k);
            acc0 = wmma_f32(a0, b0, acc0);
            v2f a1 = *(const v2f*)(arow + k + 4);
            v2f b1 = *(const v2f*)(bp + k + 4);
            acc1 = wmma_f32(a1, b1, acc1);
            v2f a2 = *(const v2f*)(arow + k + 8);
            v2f b2 = *(const v2f*)(bp + k + 8);
            acc2 = wmma_f32(a2, b2, acc2);
            v2f a3 = *(const v2f*)(arow + k + 12);
            v2f b3 = *(const v2f*)(bp + k + 12);
            acc3 = wmma_f32(a3, b3, acc3);
        }
        __syncthreads();   // all waves done reading buf (c&1) before reuse
    }

    v8f acc = (acc0 + acc1) + (acc2 + acc3);

    // Training-time expert mask (allowed_ids are unique).
    const int col = expertBase + m;
    bool allowed = false;
#pragma unroll
    for (int i = 0; i < 32; ++i) allowed |= (allowed_ids[i] == col);

    const int rowTop = rowBase + half * 8;
#pragma unroll
    for (int i = 0; i < 8; ++i) {
        float v = allowed ? acc[i] : NEG_MASK;
        logits_out[(size_t)(rowTop + i) * E_EXP + col] = v;
    }
}

// ---------------------------------------------------------------------------
// Kernel 2: per-row softmax + top-8 + renorm. One thread per token row; the
// 64 logits stay in registers (all indexing static after full unroll).
// Softmax denominator cancels under top-k renormalization:
//   w_k = exp(l_k - max) / sum_{j in top8} exp(l_j - max)
// Argmax ties resolve to the lowest index (strict >), matching lax.top_k.
// ---------------------------------------------------------------------------
__global__ __launch_bounds__(128) void softmax_topk_kernel(
    const float* __restrict__ logits,      // [T, E] masked
    float*       __restrict__ weights_out, // [T, 8]
    int*         __restrict__ experts_out) // [T, 8]
{
    const int row = blockIdx.x * blockDim.x + threadIdx.x;
    if (row >= T_TOK) return;

    float l[E_EXP];
    const float4* __restrict__ src =
        (const float4*)(logits + (size_t)row * E_EXP);
#pragma unroll
    for (int j = 0; j < E_EXP / 4; ++j) {
        float4 v = src[j];
        l[4 * j + 0] = v.x; l[4 * j + 1] = v.y;
        l[4 * j + 2] = v.z; l[4 * j + 3] = v.w;
    }

    float mx = l[0];
#pragma unroll
    for (int j = 1; j < E_EXP; ++j) mx = fmaxf(mx, l[j]);

    float tv[TOPK];
    int   ti[TOPK];
#pragma unroll
    for (int k = 0; k < TOPK; ++k) {
        float best = NEG_INF;
        int   bi   = 0;
#pragma unroll
        for (int j = 0; j < E_EXP; ++j) {
            if (l[j] > best) { best = l[j]; bi = j; }
        }
        tv[k] = best;
        ti[k] = bi;
        // static-index elimination (lowers to v_cndmask, keeps l[] in VGPRs)
#pragma unroll
        for (int j = 0; j < E_EXP; ++j) {
            if (j == bi) l[j] = NEG_INF;
        }
    }

    float esum = 0.f;
#pragma unroll
    for (int k = 0; k < TOPK; ++k) {
        tv[k] = __expf(tv[k] - mx);
        esum += tv[k];
    }
    const float inv = 1.0f / esum;

#pragma unroll
    for (int k = 0; k < TOPK; ++k) {
        weights_out[(size_t)row * TOPK + k] = tv[k] * inv;
        experts_out[(size_t)row * TOPK + k] = ti[k];
    }
}

// ---------------------------------------------------------------------------
extern "C" void kernel_launch(void* const* d_in, const int* in_sizes, int n_in,
                              void* d_out, int out_size, void* d_ws, size_t ws_size,
                              hipStream_t stream) {
    const float* hs  = (const float*)d_in[0]; // hidden_states [T, H]
    const float* wt  = (const float*)d_in[1]; // weight        [E, H]
    const int*   ids = (const int*)d_in[2];   // allowed_ids   [32]

    float* logits   = (float*)d_out;                           // [T, E]
    float* rweights = logits + (size_t)T_TOK * E_EXP;          // [T, 8]
    int*   sexperts = (int*)(rweights + (size_t)T_TOK * TOPK); // [T, 8] i32

    router_gemm_kernel<<<T_TOK / 16, 128, 0, stream>>>(hs, wt, ids, logits);
    softmax_topk_kernel<<<T_TOK / 128, 128, 0, stream>>>(logits, rweights,
                                                         sexperts);
}